// BilinearInteractionLayer_1434519077110
// MI455X (gfx1250) — compile-verified
//
#include <hip/hip_runtime.h>

// ---------------------------------------------------------------------------
// Bilinear interaction: out[b,p,e] = (sum_d v_i[b,d] * W[p,e,d]) * v_j[b,e]
// fp32 end-to-end via CDNA5 V_WMMA_F32_16X16X4_F32 (wave32).
// ---------------------------------------------------------------------------

typedef float v2f  __attribute__((ext_vector_type(2)));
typedef float v4ff __attribute__((ext_vector_type(4)));
typedef float v8f  __attribute__((ext_vector_type(8)));

#if !defined(__HIP_DEVICE_COMPILE__) || __has_builtin(__builtin_amdgcn_wmma_f32_16x16x4_f32)
#else
#error "missing __builtin_amdgcn_wmma_f32_16x16x4_f32 on this toolchain"
#endif

#define NUM_FIELDS 32
#define EMB        128
#define BATCH      2048
#define PAIRS      496          // 32*31/2
#define BTILE      128          // batch rows per workgroup (8 waves x 16)
#define LDS_STRIDE 130          // 128 + 2 pad: keeps float2 align, spreads banks
#define LDS_TILE   (EMB * LDS_STRIDE)   // floats per staged 128x128 tile

__global__ __launch_bounds__(256)
void bilinear_wmma_f32_kernel(const float* __restrict__ emb,   // [B][32][128]
                              const float* __restrict__ W,     // [496][128][128] (e,d)
                              float* __restrict__ out)         // [B][496][128]
{
    extern __shared__ float smem[];
    float* sW = smem;             // W[p] as [e][d], row stride LDS_STRIDE
    float* sV = smem + LDS_TILE;  // v_i tile as [row][d], row stride LDS_STRIDE

    const int p  = blockIdx.x;            // pair index
    const int b0 = blockIdx.y * BTILE;    // batch tile origin

    // pair p -> (fi, fj), lexicographic combinations(32, 2) order (uniform scalar)
    int fi = 0, rem = p;
    while (rem >= (NUM_FIELDS - 1 - fi)) { rem -= (NUM_FIELDS - 1 - fi); ++fi; }
    const int fj = fi + 1 + rem;

    const int tid = threadIdx.x;          // 0..255

    // ---- Stage W[p] and the v_i batch tile into LDS (float4 per thread) ----
    const float* Wp = W + (size_t)p * (EMB * EMB);
    #pragma unroll
    for (int it = 0; it < (EMB * EMB) / (256 * 4); ++it) {   // 16 iterations
        const int flat = (it * 256 + tid) * 4;               // element index
        const int row  = flat >> 7;                          // /128
        const int col  = flat & 127;
        // W[p][row=e][col..col+3]  (row length 128 divisible by 4 -> stays in row)
        v4ff w4 = *(const v4ff*)(Wp + flat);
        *(v4ff*)(sW + row * LDS_STRIDE + col) = w4;
        // v_i rows: feature_emb[b0+row][fi][col..col+3]
        v4ff a4 = *(const v4ff*)(emb + ((size_t)(b0 + row) * NUM_FIELDS + fi) * EMB + col);
        *(v4ff*)(sV + row * LDS_STRIDE + col) = a4;
    }
    __syncthreads();

    const int wave = tid >> 5;        // 0..7 (wave32)
    const int lane = tid & 31;
    const int half = lane >> 4;       // 0 / 1 (selects K pair within fragment)
    const int l15  = lane & 15;       // M (for A) / N (for B)

    // Accumulators: 8 column tiles of 16x16 f32 -> 64 VGPRs
    v8f acc[8] = {};

    // A fragment base: row (wave*16 + l15), K offset starts at 2*half
    const float* aBase = sV + (size_t)(wave * 16 + l15) * LDS_STRIDE + 2 * half;
    // B fragment base: B[k][n] = W[e = n][d = k]; lane -> e = nt*16 + l15
    const float* bBase = sW + (size_t)l15 * LDS_STRIDE + 2 * half;

    #pragma unroll 4
    for (int k = 0; k < EMB; k += 4) {
        v2f a = *(const v2f*)(aBase + k);      // A[m][k..k+1] for this half-wave
        #pragma unroll
        for (int t = 0; t < 8; ++t) {
            v2f b = *(const v2f*)(bBase + (size_t)t * 16 * LDS_STRIDE + k);
            // D = A(16x4) x B(4x16) + C ; args: neg_a,A,neg_b,B,c_mod,C,reuse_a,reuse_b
            acc[t] = __builtin_amdgcn_wmma_f32_16x16x4_f32(
                         false, a, false, b, (short)0, acc[t], false, false);
        }
    }

    // ---- Epilogue: multiply by v_j and store ----
    // C/D layout: VGPR r, lanes 0-15 -> M=r, lanes 16-31 -> M=r+8; N = lane&15
    #pragma unroll
    for (int t = 0; t < 8; ++t) {
        const int e = t * 16 + l15;
        #pragma unroll
        for (int r = 0; r < 8; ++r) {
            const int b  = b0 + wave * 16 + half * 8 + r;
            const float vj = emb[((size_t)b * NUM_FIELDS + fj) * EMB + e];
            out[((size_t)b * PAIRS + p) * EMB + e] = acc[t][r] * vj;
        }
    }
}

extern "C" void kernel_launch(void* const* d_in, const int* in_sizes, int n_in,
                              void* d_out, int out_size, void* d_ws, size_t ws_size,
                              hipStream_t stream) {
    const float* emb = (const float*)d_in[0];   // feature_emb [2048][32][128] f32
    const float* W   = (const float*)d_in[1];   // W [496][128][128] f32
    float* out       = (float*)d_out;           // [2048][496][128] f32

    (void)in_sizes; (void)n_in; (void)out_size; (void)d_ws; (void)ws_size;

    const size_t smem_bytes = 2 * (size_t)LDS_TILE * sizeof(float);  // ~130 KB (gfx1250: 320 KB/WG)
    hipFuncSetAttribute((const void*)bilinear_wmma_f32_kernel,
                        hipFuncAttributeMaxDynamicSharedMemorySize,
                        (int)smem_bytes);

    dim3 grid(PAIRS, BATCH / BTILE);  // (496, 16)
    bilinear_wmma_f32_kernel<<<grid, 256, smem_bytes, stream>>>(emb, W, out);
}